// ROI_Pool_58669253263553
// MI455X (gfx1250) — compile-verified
//
#include <hip/hip_runtime.h>
#include <stdint.h>

// ROI max-pool, N=64, C=256, H=W=56, 7x7 bins, fp32.
// Bandwidth/latency-bound (~67MB window traffic @ 23.3TB/s ~ 3us).
// Stages each (n,c) ROI window into LDS with ONE Tensor-Data-Mover
// 2D descriptor via inline asm (TENSOR_LOAD_TO_LDS, TENSORcnt) —
// portable across both toolchains' differing builtin arities.

#define PHB 7
#define PWB 7
#define NB 64
#define CC 256
#define HH 56
#define WW 56
#define WAVES_PER_BLOCK 8

typedef unsigned int v4u __attribute__((ext_vector_type(4)));
typedef unsigned int v8u __attribute__((ext_vector_type(8)));

__global__ __launch_bounds__(256) void roi_pool_kernel(
    const float* __restrict__ fm, const int* __restrict__ rois,
    float* __restrict__ out)
{
    // Per-wave scratch: up to 32x32 window (4KB) + 7x32 row-pooled tile.
    __shared__ float win[WAVES_PER_BLOCK][32 * 32];
    __shared__ float tbuf[WAVES_PER_BLOCK][PHB * 32];

    const int tid  = threadIdx.x;
    const int wave = tid >> 5;
    const int lane = tid & 31;

    const int b  = blockIdx.x;          // 0..2047
    const int n  = b >> 5;              // / (C / WAVES_PER_BLOCK) = 32
    const int cg = b & 31;
    const int c  = cg * WAVES_PER_BLOCK + wave;

    // ROI bounds — uniform across the block (same n) -> SGPR math.
    const int ry = rois[n * 4 + 0];
    const int rx = rois[n * 4 + 1];
    const int rh = rois[n * 4 + 2];
    const int rw = rois[n * 4 + 3];
    const int roh = rh >> 1;
    const int rov = rw >> 1;
    int ymin = ry - roh; if (ymin < 0) ymin = 0;
    int ymax = ry + roh; if (ymax > HH) ymax = HH - 1;   // exclusive bound
    int xmin = rx - rov; if (xmin < 0) xmin = 0;
    int xmax = rx + rov; if (xmax > WW) xmax = WW - 1;   // exclusive bound
    const int h = ymax - ymin;                           // 1..32
    const int w = xmax - xmin;                           // 1..32

    // ---- One TDM descriptor: h x w fp32 tile, row stride W, dense into LDS.
    // Wave-divergent words (lds/global addr depend on wave id) are in fact
    // wave-uniform -> readfirstlane them so they satisfy the "s" constraint.
    const unsigned ldsOffR = (unsigned)__builtin_amdgcn_readfirstlane(
        (int)(unsigned)(uintptr_t)&win[wave][0]);
    const uint64_t gaddr = (uint64_t)(uintptr_t)fm +
        4ull * (uint64_t)(((n * CC + c) * HH + ymin) * WW + xmin);
    const unsigned gloR = (unsigned)__builtin_amdgcn_readfirstlane((int)(unsigned)gaddr);
    const unsigned ghiR = (unsigned)__builtin_amdgcn_readfirstlane(
        (int)((unsigned)(gaddr >> 32) & 0x01FFFFFFu));

    // D# group 0 (4 SGPRs): count=1 | lds_addr | global_addr[56:0] | type=2
    v4u g0 = { 1u, ldsOffR, gloR, ghiR | (2u << 30) };
    // D# group 1 (8 SGPRs): wg_mask=0 | data_size=2 (4B)
    //   tensor_dim0=w @ bits 79:48   tensor_dim1=h @ bits 111:80
    //   tile_dim0=w  @ bits 127:112  tile_dim1=h @ bits 143:128
    //   tensor_dim0_stride=W @ bits 207:160
    v8u g1 = { 0x00020000u,
               (unsigned)w << 16,
               (unsigned)h << 16,
               (unsigned)w << 16,
               (unsigned)h,
               (unsigned)WW,
               0u, 0u };
    // 2D tensor: VADDR2/VADDR3 omitted (NULL) -> 2-group descriptor form.
    asm volatile("tensor_load_to_lds %0, %1"
                 :: "s"(g0), "s"(g1)
                 : "memory");

    const int row_stride = w;               // TDM packs rows densely

    // Row-bin bounds (uniform). re uses ceil; bins may overlap by one row.
    int rs[PHB], re[PHB];
#pragma unroll
    for (int ph = 0; ph < PHB; ++ph) {
        rs[ph] = ymin + (ph * h) / PHB;
        re[ph] = ymin + ((ph + 1) * h + PHB - 1) / PHB;
    }

    asm volatile("s_wait_tensorcnt 0" ::: "memory");

    // ---- Phase 1: row pool from LDS. lane = window column. ----
    float m[PHB];
#pragma unroll
    for (int ph = 0; ph < PHB; ++ph) m[ph] = -INFINITY;
    int rowbase = 0;
    for (int y = ymin; y < ymax; ++y) {
        float v = win[wave][rowbase + lane];  // consecutive addrs: conflict-free
#pragma unroll
        for (int ph = 0; ph < PHB; ++ph)
            if (y >= rs[ph] && y < re[ph]) m[ph] = fmaxf(m[ph], v);
        rowbase += row_stride;
    }
#pragma unroll
    for (int ph = 0; ph < PHB; ++ph) tbuf[wave][ph * 32 + lane] = m[ph];
    // (lanes >= w produced garbage columns; phase 2 never reads them)

    // ---- Phase 2: column pool. lane = bin (strided over 49 bins). ----
    for (int bin = lane; bin < PHB * PWB; bin += 32) {
        const int ph  = bin / PWB;
        const int pw  = bin - ph * PWB;
        const int csr = (pw * w) / PWB;                      // relative to xmin
        const int cer = ((pw + 1) * w + PWB - 1) / PWB;      // ceil, <= w
        float mx = -INFINITY;
        for (int xx = csr; xx < cer; ++xx)
            mx = fmaxf(mx, tbuf[wave][ph * 32 + xx]);
        out[((n * CC + c) * PHB + ph) * PWB + pw] = mx;
    }
}

extern "C" void kernel_launch(void* const* d_in, const int* in_sizes, int n_in,
                              void* d_out, int out_size, void* d_ws, size_t ws_size,
                              hipStream_t stream) {
    (void)in_sizes; (void)n_in; (void)out_size; (void)d_ws; (void)ws_size;
    const float* fm   = (const float*)d_in[0];
    const int*   rois = (const int*)d_in[1];
    float*       out  = (float*)d_out;
    // One wave per (n, c): 64 * (256/8) = 2048 blocks of 256 threads (8 waves).
    roi_pool_kernel<<<dim3(NB * (CC / WAVES_PER_BLOCK)), dim3(256), 0, stream>>>(fm, rois, out);
}